// ModelLocal_86912958202187
// MI455X (gfx1250) — compile-verified
//
#include <hip/hip_runtime.h>
#include <math.h>

typedef float v2f __attribute__((ext_vector_type(2)));
typedef float v8f __attribute__((ext_vector_type(8)));

#define M_TOT  8192
#define W_CTX  100
#define K_CAND 6
#define D_DIM  300
#define R_TOP  25
#define H_MLP  100

// LDS layout in floats (all float4-aligned where vector-accessed)
constexpr int LDS_CTXT  = 0;        // 100*300 = 30000
constexpr int LDS_CANDA = 30000;    // 16*300  = 4800  (cand*A, zero-padded k>=6)
constexpr int LDS_CAND  = 34800;    // 6*300   = 1800  (raw cand, for sim)
constexpr int LDS_A     = 36600;    // 300
constexpr int LDS_B     = 36900;    // 300
constexpr int LDS_UMAT  = 37200;    // 100*6   = 600
constexpr int LDS_U     = 37800;    // 100
constexpr int LDS_RED   = 37904;    // 256
constexpr int LDS_BETA  = 38160;    // 100
constexpr int LDS_CF    = 38260;    // 300
constexpr int LDS_SCAL  = 38560;    // 8
constexpr int LDS_TOTAL = 38568;    // 154,272 B -> 2 workgroups / WGP

__global__ __launch_bounds__(256) void entity_link_fused(
    const float* __restrict__ ctxt_g,   // [M,W,D]
    const float* __restrict__ cand_g,   // [M,K,D]
    const float* __restrict__ pem_g,    // [M,K]
    const float* __restrict__ A_g,      // [D]
    const float* __restrict__ B_g,      // [D]
    const float* __restrict__ W1_g,     // [H,2]
    const float* __restrict__ b1_g,     // [H]
    const float* __restrict__ W2_g,     // [H]
    const float* __restrict__ b2_g,     // [1]
    float* __restrict__ out_x,          // [M,K]
    float* __restrict__ out_beta,       // [M,W]
    float* __restrict__ out_sim)        // [M,K]
{
    extern __shared__ float lds[];
    const int m    = blockIdx.x;
    const int tid  = threadIdx.x;
    const int lane = tid & 31;
    const int wave = tid >> 5;

    // ---------- Phase A: stage ctxt / cand / A / B into LDS (b128 loads) ----
    {
        const float4* src = (const float4*)(ctxt_g + (size_t)m * (W_CTX * D_DIM));
        float4* dst = (float4*)(lds + LDS_CTXT);
        for (int i = tid; i < (W_CTX * D_DIM) / 4; i += 256) dst[i] = src[i];
    }
    {
        const float4* src = (const float4*)(cand_g + (size_t)m * (K_CAND * D_DIM));
        float4* dst = (float4*)(lds + LDS_CAND);
        for (int i = tid; i < (K_CAND * D_DIM) / 4; i += 256) dst[i] = src[i];
    }
    {
        const float4* sa = (const float4*)A_g; float4* da = (float4*)(lds + LDS_A);
        const float4* sb = (const float4*)B_g; float4* db = (float4*)(lds + LDS_B);
        for (int i = tid; i < D_DIM / 4; i += 256) { da[i] = sa[i]; db[i] = sb[i]; }
    }
    __syncthreads();

    // ---------- Phase A2: candA = cand * A, zero-padded to 16 rows ----------
    // Hoists the per-iteration pre_u scaling and k-mask out of the WMMA loop.
    {
        float4* dst = (float4*)(lds + LDS_CANDA);
        const float4* cnd = (const float4*)(lds + LDS_CAND);
        const float4* av  = (const float4*)(lds + LDS_A);
        constexpr int ROWQ = D_DIM / 4;                    // 75 float4 per row
        for (int i = tid; i < 16 * ROWQ; i += 256) {
            int k = i / ROWQ, dq = i - k * ROWQ;
            float4 v = make_float4(0.f, 0.f, 0.f, 0.f);
            if (k < K_CAND) {
                float4 c = cnd[k * ROWQ + dq];
                float4 a = av[dq];
                v = make_float4(c.x * a.x, c.y * a.y, c.z * a.z, c.w * a.w);
            }
            dst[i] = v;
        }
    }
    __syncthreads();

    // ---------- Phase B: u-partials via V_WMMA_F32_16X16X4_F32 --------------
    // Output tile: rows = 16 context words (w), cols = 16 candidates (6 used,
    // rest fold in zeros from padded candA).  One w-tile per wave (7 tiles).
    if (wave < 7) {
        const int wbase = wave * 16;
        const int koff  = (lane < 16) ? 0 : 2;       // fp32 A/B fragment layout
        const int arow  = wbase + (lane & 15);
        const int kcol  = lane & 15;
        const float* brow = lds + LDS_CANDA + kcol * D_DIM;
        v8f acc = {};

        if (wave < 6) {
            // clean path: all 16 w-rows valid, no masking in the loop
            const float* arowp = lds + LDS_CTXT + arow * D_DIM;
            for (int d0 = 0; d0 < D_DIM; d0 += 4) {
                v2f a, b;
                a.x = arowp[d0 + koff];
                a.y = arowp[d0 + koff + 1];
                b.x = brow[d0 + koff];
                b.y = brow[d0 + koff + 1];
                acc = __builtin_amdgcn_wmma_f32_16x16x4_f32(
                    false, a, false, b, (short)0, acc, false, false);
            }
        } else {
            // tail tile (w 96..111): mask rows >= 100 with zeros
            const bool rowok = (arow < W_CTX);
            const float* arowp = lds + LDS_CTXT + (rowok ? arow : 0) * D_DIM;
            for (int d0 = 0; d0 < D_DIM; d0 += 4) {
                v2f a, b;
                float a0 = arowp[d0 + koff];
                float a1 = arowp[d0 + koff + 1];
                a.x = rowok ? a0 : 0.f;
                a.y = rowok ? a1 : 0.f;
                b.x = brow[d0 + koff];
                b.y = brow[d0 + koff + 1];
                acc = __builtin_amdgcn_wmma_f32_16x16x4_f32(
                    false, a, false, b, (short)0, acc, false, false);
            }
        }
        // Scatter C tile: VGPR j, lanes 0-15 -> (w = wbase+j, k = lane),
        //                  lanes 16-31 -> (w = wbase+j+8, k = lane-16)
        for (int j = 0; j < 8; ++j) {
            int wr = wbase + j + ((lane < 16) ? 0 : 8);
            int kk = lane & 15;
            if (wr < W_CTX && kk < K_CAND)
                lds[LDS_UMAT + wr * K_CAND + kk] = acc[j];
        }
    }
    __syncthreads();

    // ---------- Phase C: u[w] = max_k ---------------------------------------
    float myu = -1e30f;
    if (tid < W_CTX) {
        float mx = lds[LDS_UMAT + tid * K_CAND];
        for (int k = 1; k < K_CAND; ++k)
            mx = fmaxf(mx, lds[LDS_UMAT + tid * K_CAND + k]);
        lds[LDS_U + tid] = mx;
        myu = mx;
    }
    __syncthreads();

    // ---------- Phase D: exact top-R threshold (rank count) + softmax -------
    if (tid < W_CTX) {
        int cg = 0, ce = 0;
        for (int j = 0; j < W_CTX; ++j) {
            float v = lds[LDS_U + j];
            cg += (v > myu);
            ce += (v == myu);
        }
        // 25th largest (sorted[24]): unique value v with #{>v} <= 24 < #{>=v}
        if (cg <= (R_TOP - 1) && (cg + ce) > (R_TOP - 1))
            lds[LDS_SCAL + 0] = myu;
    }
    __syncthreads();
    float top_min = lds[LDS_SCAL + 0];
    float mk = (tid < W_CTX)
                   ? ((myu - top_min > 0.f) ? myu : top_min - 50.f)
                   : -1e30f;
    // block max
    lds[LDS_RED + tid] = mk;
    __syncthreads();
    for (int s = 128; s > 0; s >>= 1) {
        if (tid < s) lds[LDS_RED + tid] = fmaxf(lds[LDS_RED + tid], lds[LDS_RED + tid + s]);
        __syncthreads();
    }
    float rmax = lds[LDS_RED + 0];
    __syncthreads();
    float e = (tid < W_CTX) ? __expf(mk - rmax) : 0.f;
    lds[LDS_RED + tid] = e;
    __syncthreads();
    for (int s = 128; s > 0; s >>= 1) {
        if (tid < s) lds[LDS_RED + tid] += lds[LDS_RED + tid + s];
        __syncthreads();
    }
    float denom = lds[LDS_RED + 0];
    float beta = e / denom;
    if (tid < W_CTX) {
        lds[LDS_BETA + tid] = beta;
        out_beta[(size_t)m * W_CTX + tid] = beta;
    }
    __syncthreads();

    // ---------- Phase E: ctxt_full[d] = sum_w beta[w]*ctxt[w,d], scale by B -
    for (int d = tid; d < D_DIM; d += 256) {
        float acc = 0.f;
        for (int w = 0; w < W_CTX; ++w)
            acc = fmaf(lds[LDS_BETA + w], lds[LDS_CTXT + w * D_DIM + d], acc);
        lds[LDS_CF + d] = acc * lds[LDS_B + d];
    }
    __syncthreads();

    // ---------- Phase F+G: sim[k] and MLP scorer, one wave per candidate ----
    if (wave < K_CAND) {
        const int k = wave;
        float p = 0.f;
        for (int d = lane; d < D_DIM; d += 32)
            p = fmaf(lds[LDS_CAND + k * D_DIM + d], lds[LDS_CF + d], p);
        for (int off = 16; off; off >>= 1) p += __shfl_xor(p, off, 32);
        // butterfly: every lane now holds sim
        if (lane == 0) out_sim[(size_t)m * K_CAND + k] = p;

        float pe = pem_g[(size_t)m * K_CAND + k];
        float acc2 = 0.f;
        for (int h = lane; h < H_MLP; h += 32) {
            float hv = fmaf(W1_g[h * 2 + 0], p,
                       fmaf(W1_g[h * 2 + 1], pe, b1_g[h]));
            hv = fmaxf(hv, 0.f);
            acc2 = fmaf(W2_g[h], hv, acc2);
        }
        for (int off = 16; off; off >>= 1) acc2 += __shfl_xor(acc2, off, 32);
        if (lane == 0) out_x[(size_t)m * K_CAND + k] = acc2 + b2_g[0];
    }
}

extern "C" void kernel_launch(void* const* d_in, const int* in_sizes, int n_in,
                              void* d_out, int out_size, void* d_ws, size_t ws_size,
                              hipStream_t stream) {
    const float* ctxt = (const float*)d_in[0];
    const float* cand = (const float*)d_in[1];
    const float* pem  = (const float*)d_in[2];
    const float* A    = (const float*)d_in[3];
    const float* B    = (const float*)d_in[4];
    const float* W1   = (const float*)d_in[5];
    const float* b1   = (const float*)d_in[6];
    const float* W2   = (const float*)d_in[7];
    const float* b2   = (const float*)d_in[8];

    float* out      = (float*)d_out;
    float* out_x    = out;                                    // [M,K]
    float* out_beta = out + (size_t)M_TOT * K_CAND;           // [M,W]
    float* out_sim  = out + (size_t)M_TOT * (K_CAND + W_CTX); // [M,K]

    size_t shmem = (size_t)LDS_TOTAL * sizeof(float);         // ~154 KB
    entity_link_fused<<<M_TOT, 256, shmem, stream>>>(
        ctxt, cand, pem, A, B, W1, b1, W2, b2, out_x, out_beta, out_sim);
}